// DistanceTransform_11647951307554
// MI455X (gfx1250) — compile-verified
//
#include <hip/hip_runtime.h>
#include <math.h>
#include <stdint.h>

typedef float v2f __attribute__((ext_vector_type(2)));
typedef float v8f __attribute__((ext_vector_type(8)));

#define HPAR 0.35f
#define PITCH 130            // 128 + 2 pad floats: keeps column & row-pair LDS access low-conflict
#define NITERS 128           // ceil(max(H,W) / (ks//2)) = 128

__launch_bounds__(256, 1)
__global__ void dt_kernel(const float* __restrict__ img, float* __restrict__ out) {
    // Whole image resident in LDS: boundary field + C1 (vertical [b,a,b] conv) scratch.
    __shared__ float bnd[128 * PITCH];
    __shared__ float c1s[128 * PITCH];
    __shared__ int   flag;

    const int tid   = threadIdx.x;
    const int lane  = tid & 31;
    const int wave  = tid >> 5;          // 8 waves of 32 (wave32)
    const int imgid = blockIdx.x;
    const float* src = img + (size_t)imgid * (128 * 128);

    // kernel weights: a = exp(-1/h), b = exp(-sqrt2/h)  (constant-folded)
    const float wA = expf(-1.0f / 0.35f);
    const float wB = expf(-sqrtf(2.0f) / 0.35f);

    // ---- Precompute banded A matrices for V_WMMA_F32_16X16X4_F32 ----
    // A 16x4 layout: lanes 0-15 -> M=lane (K = v + 0), lanes 16-31 -> M=lane-16 (K = v + 2)
    // T[m, j] = w[j - m] for j-m in {0,1,2}, else 0  (j = extended row index, 0..17, padded to 20)
    const int mrow  = lane & 15;
    const int khalf = (lane >> 4) << 1;  // 0 or 2
    float A0[5][2], A1[5][2];            // C0 kernel [a,1,a]; C1 kernel [b,a,b]
#pragma unroll
    for (int c = 0; c < 5; ++c) {
#pragma unroll
        for (int v = 0; v < 2; ++v) {
            int kk = 4 * c + v + khalf;
            int d  = kk - mrow;
            A0[c][v] = (d == 0) ? wA : (d == 1) ? 1.0f : (d == 2) ? wA : 0.0f;
            A1[c][v] = (d == 0) ? wB : (d == 1) ? wA   : (d == 2) ? wB : 0.0f;
        }
    }

    // ---- Async copy image -> LDS boundary field (global_load_async_to_lds_b128) ----
    // Each wave copies one 128-float row per issue (32 lanes x 16B), rows strided by 8.
    for (int r = wave; r < 128; r += 8) {
        unsigned ldsoff = (unsigned)(uintptr_t)(&bnd[r * PITCH + lane * 4]);
        unsigned goff   = (unsigned)((r * 128 + lane * 4) * 4);
        asm volatile("global_load_async_to_lds_b128 %0, %1, %2 offset:0"
                     :: "v"(ldsoff), "v"(goff), "s"(src) : "memory");
    }
    if (tid == 0) flag = 0;
    asm volatile("s_wait_asynccnt 0" ::: "memory");
    __syncthreads();

    // out accumulator lives in registers: 8 tiles/wave x 8 rows/lane
    float acc[8][8];
#pragma unroll
    for (int k = 0; k < 8; ++k)
#pragma unroll
        for (int r = 0; r < 8; ++r) acc[k][r] = 0.0f;

    // Each wave owns tiles t = wave*8 + k over the 8x8 grid of 16x16 tiles.
    for (int it = 0; it < NITERS; ++it) {
        // ---------- Stage 1: C1 = [b,a,b] vertical conv, whole image -> LDS ----------
        for (int k = 0; k < 8; ++k) {
            int t  = wave * 8 + k;
            int y0 = (t >> 3) << 4, x0 = (t & 7) << 4;
            v8f d = {0.f, 0.f, 0.f, 0.f, 0.f, 0.f, 0.f, 0.f};
#pragma unroll
            for (int c = 0; c < 5; ++c) {
                v2f a = {A1[c][0], A1[c][1]};
                // B 4x16 layout: lane->N (=mrow), vgpr+half->K; row clamp = replicate pad (y)
                int r0  = y0 - 1 + 4 * c + khalf;
                int rr0 = min(max(r0, 0), 127);
                int rr1 = min(max(r0 + 1, 0), 127);
                int xx  = x0 + mrow;
                v2f b = {bnd[rr0 * PITCH + xx], bnd[rr1 * PITCH + xx]};
                d = __builtin_amdgcn_wmma_f32_16x16x4_f32(false, a, false, b,
                                                          (short)0, d, false, false);
            }
            // D layout: lane->N, vgpr r -> M = r + (lane>=16 ? 8 : 0)
            int ybase = y0 + ((lane >> 4) << 3);
            int xx    = x0 + mrow;
#pragma unroll
            for (int r = 0; r < 8; ++r) c1s[(ybase + r) * PITCH + xx] = d[r];
        }
        __syncthreads();

        // ---------- Stage 2: C0 + horizontal combine + log + update ----------
        bool any = false;
        for (int k = 0; k < 8; ++k) {
            int t  = wave * 8 + k;
            int y0 = (t >> 3) << 4, x0 = (t & 7) << 4;
            v8f d = {0.f, 0.f, 0.f, 0.f, 0.f, 0.f, 0.f, 0.f};
#pragma unroll
            for (int c = 0; c < 5; ++c) {
                v2f a = {A0[c][0], A0[c][1]};
                int r0  = y0 - 1 + 4 * c + khalf;
                int rr0 = min(max(r0, 0), 127);
                int rr1 = min(max(r0 + 1, 0), 127);
                int xx  = x0 + mrow;
                v2f b = {bnd[rr0 * PITCH + xx], bnd[rr1 * PITCH + xx]};
                d = __builtin_amdgcn_wmma_f32_16x16x4_f32(false, a, false, b,
                                                          (short)0, d, false, false);
            }
            int ybase = y0 + ((lane >> 4) << 3);
            int xx    = x0 + mrow;
            int xl    = max(xx - 1, 0);      // replicate pad (x)
            int xr    = min(xx + 1, 127);
#pragma unroll
            for (int r = 0; r < 8; ++r) {
                int   y    = ybase + r;
                float conv = d[r] + c1s[y * PITCH + xl] + c1s[y * PITCH + xr];
                if (conv > 0.0f) {                       // conv==0 -> +inf -> nan_to_num -> 0
                    float cdt = -HPAR * __logf(conv);
                    if (cdt > 0.0f) {                    // mask
                        acc[k][r] += (float)it + cdt;    // offset = it * (ks//2) = it
                        bnd[y * PITCH + xx] = 1.0f;      // boundary := 1 (monotone)
                        any = true;
                    }
                }
            }
        }
        if (any) flag = 1;
        __syncthreads();                  // bnd + flag stable
        int done = (flag == 0);
        __syncthreads();                  // everyone has read flag
        if (tid == 0) flag = 0;
        if (done) break;                  // uniform early exit: remaining iters contribute 0
    }

    // ---- Write register-resident result to global ----
    float* dst = out + (size_t)imgid * (128 * 128);
#pragma unroll
    for (int k = 0; k < 8; ++k) {
        int t  = wave * 8 + k;
        int y0 = (t >> 3) << 4, x0 = (t & 7) << 4;
        int ybase = y0 + ((lane >> 4) << 3);
        int xx    = x0 + mrow;
#pragma unroll
        for (int r = 0; r < 8; ++r) dst[(ybase + r) * 128 + xx] = acc[k][r];
    }
}

extern "C" void kernel_launch(void* const* d_in, const int* in_sizes, int n_in,
                              void* d_out, int out_size, void* d_ws, size_t ws_size,
                              hipStream_t stream) {
    const float* img = (const float*)d_in[0];
    float* out = (float*)d_out;
    int n_images = in_sizes[0] / (128 * 128);   // B*C = 24
    dt_kernel<<<n_images, 256, 0, stream>>>(img, out);
}